// Classifier_30382598652208
// MI455X (gfx1250) — compile-verified
//
#include <hip/hip_runtime.h>
#include <hip/hip_bf16.h>

typedef float v2f __attribute__((ext_vector_type(2)));
typedef float v8f __attribute__((ext_vector_type(8)));

#define BATCH 16
#define S_LEN 1001
#define S_PAD 1008      // 63 tiles of 16
#define NTILES_S 63
#define D_DIM 512
#define BAND 50
#define QS_STRIDE 516   // 512 + 4 pad -> conflict-free WMMA A reads from LDS
#define SC_STRIDE 1012  // 1008 + 4 pad

__device__ __forceinline__ v8f wmma_f32_4(v2f a, v2f b, v8f c) {
    // V_WMMA_F32_16X16X4_F32: D = A(16x4,f32) * B(4x16,f32) + C(16x16,f32)
    return __builtin_amdgcn_wmma_f32_16x16x4_f32(
        /*neg_a=*/false, a, /*neg_b=*/false, b,
        /*c_mod=*/(short)0, c, /*reuse_a=*/false, /*reuse_b=*/false);
}

// ---------------------------------------------------------------------------
// Kernel 1: fused Q/K/V projection.  out[z] = x @ W[z] + b[z], padded to S_PAD.
// Two 16-row tiles per block: every weight (B) fragment feeds two WMMAs.
// grid = (504 tile-pairs, 8 n-blocks of 64, 3 matrices); block = 128 (4 waves)
// ---------------------------------------------------------------------------
__global__ __launch_bounds__(128) void qkv_kernel(
    const float* __restrict__ x,
    const float* __restrict__ Wq, const float* __restrict__ bq,
    const float* __restrict__ Wk, const float* __restrict__ bk,
    const float* __restrict__ Wv, const float* __restrict__ bv,
    float* __restrict__ qkv_out)
{
    __shared__ float xs[32 * QS_STRIDE];    // two row tiles, 66KB

    const int tp = blockIdx.x;              // row-tile pair 0..503
    const int z  = blockIdx.z;
    const float* W  = (z == 0) ? Wq : ((z == 1) ? Wk : Wv);
    const float* bs = (z == 0) ? bq : ((z == 1) ? bk : bv);
    float* O = qkv_out + (size_t)z * ((size_t)BATCH * S_PAD * D_DIM);

    const int tid = threadIdx.x;

    // stage x rows for both tiles [32 x 512] (float4 coalesced); tiles may
    // straddle a batch boundary, so map (b,s) per sub-tile.
    for (int i = tid; i < 32 * 128; i += 128) {
        const int row  = i >> 7;            // 0..31
        const int c4   = i & 127;
        const int tile = tp * 2 + (row >> 4);
        const int b    = tile / NTILES_S;
        int s = (tile % NTILES_S) * 16 + (row & 15);
        if (s > S_LEN - 1) s = S_LEN - 1;   // clamp pad rows
        const float4 v = ((const float4*)(x + ((size_t)b * S_LEN + s) * D_DIM))[c4];
        float* dst = &xs[row * QS_STRIDE + c4 * 4];
        dst[0] = v.x; dst[1] = v.y; dst[2] = v.z; dst[3] = v.w;
    }
    __syncthreads();

    const int wave = tid >> 5;
    const int lane = tid & 31;
    const int half = lane >> 4;
    const int ln   = lane & 15;
    const int n0   = blockIdx.y * 64 + wave * 16;

    v8f acc0 = {0.f, 0.f, 0.f, 0.f, 0.f, 0.f, 0.f, 0.f};
    v8f acc1 = acc0;
    const float* wcol = W + n0 + ln;        // W row-major [K=512][N=512]
    #pragma unroll 4
    for (int k0 = 0; k0 < D_DIM; k0 += 4) {
        const int kb = k0 + 2 * half;
        const v2f a0 = *(const v2f*)&xs[ln * QS_STRIDE + kb];
        const v2f a1 = *(const v2f*)&xs[(16 + ln) * QS_STRIDE + kb];
        v2f bf;
        bf.x = wcol[(size_t)kb * D_DIM];
        bf.y = wcol[(size_t)(kb + 1) * D_DIM];
        acc0 = wmma_f32_4(a0, bf, acc0);    // B fragment reused twice
        acc1 = wmma_f32_4(a1, bf, acc1);
    }

    const float bias = bs[n0 + ln];
    #pragma unroll
    for (int t = 0; t < 2; ++t) {
        const int tile = tp * 2 + t;
        const int b    = tile / NTILES_S;
        const int r0   = (tile % NTILES_S) * 16;
        float* obase = O + ((size_t)b * S_PAD + r0) * D_DIM + n0 + ln;
        const v8f acc = t ? acc1 : acc0;
        #pragma unroll
        for (int v = 0; v < 8; ++v)
            obase[(size_t)(v + 8 * half) * D_DIM] = acc[v] + bias;
    }
}

// ---------------------------------------------------------------------------
// Kernel 2: per (batch, 16-row query tile): full-row softmax, post-softmax band
// mask, L2 renorm (folded into per-row scalar), attn writeback, attn @ V.
// grid = (63, 16); block = 256 (8 waves)
// ---------------------------------------------------------------------------
__global__ __launch_bounds__(256) void attn_kernel(
    const float* __restrict__ Q, const float* __restrict__ K,
    const float* __restrict__ V,
    float* __restrict__ out_feat,   // [B,S,D]
    float* __restrict__ out_attn)   // [B,S,S]
{
    __shared__ float qs[16 * QS_STRIDE];
    __shared__ float sc[16 * SC_STRIDE];
    __shared__ float red[16 * 16];
    __shared__ float rowstat[16];
    __shared__ float rowinv[16];

    const int st = blockIdx.x;
    const int b  = blockIdx.y;
    const int r0 = st * 16;
    const int tid  = threadIdx.x;
    const int wave = tid >> 5;
    const int lane = tid & 31;
    const int half = lane >> 4;
    const int ln   = lane & 15;

    const float* Qb = Q + (size_t)b * S_PAD * D_DIM;
    const float* Kb = K + (size_t)b * S_PAD * D_DIM;
    const float* Vb = V + (size_t)b * S_PAD * D_DIM;

    // stage Q tile
    for (int i = tid; i < 16 * 128; i += 256) {
        const int row = i >> 7;
        const int c4  = i & 127;
        const float4 v = ((const float4*)(Qb + (size_t)(r0 + row) * D_DIM))[c4];
        float* dst = &qs[row * QS_STRIDE + c4 * 4];
        dst[0] = v.x; dst[1] = v.y; dst[2] = v.z; dst[3] = v.w;
    }
    __syncthreads();

    // ---- scores = Q Ktile^T / sqrt(512); two key tiles per wave pass so the
    //      shared Q (A) fragment feeds two WMMAs per LDS read ----
    const float scale = 0.044194173824159216f;   // 1/sqrt(512)
    for (int base = 0; base < NTILES_S; base += 16) {
        const int ktA  = base + wave;                    // <= 55, always valid
        const int ktB  = base + wave + 8;                // may be 63 (invalid)
        const bool hasB = (ktB < NTILES_S);
        const int ktBc = hasB ? ktB : (NTILES_S - 1);    // clamp, store-guarded
        v8f accA = {0.f, 0.f, 0.f, 0.f, 0.f, 0.f, 0.f, 0.f};
        v8f accB = accA;
        const float* krowA = Kb + (size_t)(ktA  * 16 + ln) * D_DIM;
        const float* krowB = Kb + (size_t)(ktBc * 16 + ln) * D_DIM;
        __builtin_prefetch(krowA, 0, 1);
        __builtin_prefetch(krowB, 0, 1);
        #pragma unroll 4
        for (int k0 = 0; k0 < D_DIM; k0 += 4) {
            const int kb = k0 + 2 * half;
            const v2f a  = *(const v2f*)&qs[ln * QS_STRIDE + kb];
            const v2f bA = *(const v2f*)(krowA + kb);    // B[k][n] = Key[n][k]
            const v2f bB = *(const v2f*)(krowB + kb);
            accA = wmma_f32_4(a, bA, accA);
            accB = wmma_f32_4(a, bB, accB);
        }
        #pragma unroll
        for (int v = 0; v < 8; ++v) {
            sc[(v + 8 * half) * SC_STRIDE + ktA * 16 + ln] = accA[v] * scale;
            if (hasB)
                sc[(v + 8 * half) * SC_STRIDE + ktB * 16 + ln] = accB[v] * scale;
        }
    }
    __syncthreads();

    // ---- softmax over full row (1001 valid cols), then band mask + L2 ----
    const int r = tid & 15;       // row within tile
    const int g = tid >> 4;       // 16 threads per row
    const int srow = r0 + r;      // global query index

    // pass 1: row max (padded cols -> -inf)
    float m = -3.0e38f;
    for (int c = g; c < S_PAD; c += 16) {
        float v = (c < S_LEN) ? sc[r * SC_STRIDE + c] : -3.0e38f;
        m = fmaxf(m, v);
    }
    red[r * 16 + g] = m;
    __syncthreads();
    if (tid < 16) {
        float mm = red[tid * 16];
        #pragma unroll
        for (int j = 1; j < 16; ++j) mm = fmaxf(mm, red[tid * 16 + j]);
        rowstat[tid] = mm;
    }
    __syncthreads();

    // pass 2: e = exp(s - max), sum; padded cols -> 0
    const float rmax = rowstat[r];
    float psum = 0.f;
    for (int c = g; c < S_PAD; c += 16) {
        float e = (c < S_LEN) ? __expf(sc[r * SC_STRIDE + c] - rmax) : 0.f;
        sc[r * SC_STRIDE + c] = e;
        psum += e;
    }
    red[r * 16 + g] = psum;
    __syncthreads();
    if (tid < 16) {
        float ss = 0.f;
        #pragma unroll
        for (int j = 0; j < 16; ++j) ss += red[tid * 16 + j];
        rowstat[tid] = ss;
    }
    __syncthreads();

    // pass 3: attn = e/sum, post-softmax band mask, accumulate sum of squares
    const float rsum_inv = 1.0f / rowstat[r];
    float psq = 0.f;
    for (int c = g; c < S_PAD; c += 16) {
        float v = sc[r * SC_STRIDE + c] * rsum_inv;
        int d = c - srow; if (d < 0) d = -d;
        v = (d <= BAND && c < S_LEN) ? v : 0.f;
        sc[r * SC_STRIDE + c] = v;
        psq += v * v;
    }
    red[r * 16 + g] = psq;
    __syncthreads();
    if (tid < 16) {
        float ss = 0.f;
        #pragma unroll
        for (int j = 0; j < 16; ++j) ss += red[tid * 16 + j];
        rowinv[tid] = 1.0f / fmaxf(sqrtf(ss), 1e-12f);   // F.normalize eps clamp
    }
    __syncthreads();

    // ---- write attn_final rows (full 1001 cols, coalesced; zeros included) ----
    for (int rr = 0; rr < 16; ++rr) {
        const int s = r0 + rr;
        if (s >= S_LEN) break;
        const float sc_row_inv = rowinv[rr];
        float* orow = out_attn + ((size_t)b * S_LEN + s) * S_LEN;
        for (int c = tid; c < S_LEN; c += 256)
            orow[c] = sc[rr * SC_STRIDE + c] * sc_row_inv;
    }

    // ---- compound = attn_final @ V over the banded key window only; two
    //      N-tiles per wave so the attn (A) fragment feeds two WMMAs ----
    int clo = r0 - BAND;            if (clo < 0) clo = 0;
    int chi = r0 + 15 + BAND;       if (chi > S_LEN - 1) chi = S_LEN - 1;
    const int ct_lo = clo >> 4;
    const int ct_hi = chi >> 4;     // <= 62

    for (int base = 0; base < 32; base += 16) {
        const int n0A = (base + wave) * 16;
        const int n0B = (base + wave + 8) * 16;
        v8f accA = {0.f, 0.f, 0.f, 0.f, 0.f, 0.f, 0.f, 0.f};
        v8f accB = accA;
        for (int ct = ct_lo; ct <= ct_hi; ++ct) {
            #pragma unroll
            for (int k0 = 0; k0 < 16; k0 += 4) {
                const int kabs = ct * 16 + k0 + 2 * half;
                const v2f a = *(const v2f*)&sc[ln * SC_STRIDE + kabs];
                v2f bA, bB;
                bA.x = Vb[(size_t)kabs * D_DIM + n0A + ln];
                bA.y = Vb[(size_t)(kabs + 1) * D_DIM + n0A + ln];
                bB.x = Vb[(size_t)kabs * D_DIM + n0B + ln];
                bB.y = Vb[(size_t)(kabs + 1) * D_DIM + n0B + ln];
                accA = wmma_f32_4(a, bA, accA);
                accB = wmma_f32_4(a, bB, accB);
            }
        }
        #pragma unroll
        for (int v = 0; v < 8; ++v) {
            const int mrow = v + 8 * half;
            const int s = r0 + mrow;
            if (s < S_LEN) {
                const float sc_row = rowinv[mrow];
                float* orow = out_feat + ((size_t)b * S_LEN + s) * D_DIM;
                orow[n0A + ln] = accA[v] * sc_row;
                orow[n0B + ln] = accB[v] * sc_row;
            }
        }
    }
}

// ---------------------------------------------------------------------------
extern "C" void kernel_launch(void* const* d_in, const int* in_sizes, int n_in,
                              void* d_out, int out_size, void* d_ws, size_t ws_size,
                              hipStream_t stream) {
    const float* x  = (const float*)d_in[0];
    const float* Wq = (const float*)d_in[1];
    const float* bq = (const float*)d_in[2];
    const float* Wk = (const float*)d_in[3];
    const float* bk = (const float*)d_in[4];
    const float* Wv = (const float*)d_in[5];
    const float* bv = (const float*)d_in[6];
    // d_in[7] = local_mat (band mask computed analytically on device)

    float* out_feat = (float*)d_out;                                     // [16,1001,512]
    float* out_attn = (float*)d_out + (size_t)BATCH * S_LEN * D_DIM;     // [16,1001,1001]

    const size_t mat_elems = (size_t)BATCH * S_PAD * D_DIM;
    float* Q = (float*)d_ws;
    float* K = Q + mat_elems;
    float* V = K + mat_elems;

    dim3 g1(BATCH * NTILES_S / 2, 8, 3);   // 504 row-tile pairs x 8 n-blocks x {Q,K,V}
    qkv_kernel<<<g1, 128, 0, stream>>>(x, Wq, bq, Wk, bk, Wv, bv, Q);

    dim3 g2(NTILES_S, BATCH);
    attn_kernel<<<g2, 256, 0, stream>>>(Q, K, V, out_feat, out_attn);
}